// KalmanFilter_2783138808101
// MI455X (gfx1250) — compile-verified
//
#include <hip/hip_runtime.h>

// Kalman filter scan for MI455X (gfx1250, wave32).
// One wave per batch element; all matrix work on v_wmma_f32_16x16x4_f32.

typedef __attribute__((ext_vector_type(2))) float v2f;
typedef __attribute__((ext_vector_type(8))) float v8f;

#define KF_B 512
#define KF_T 200
#define KF_S 32
#define KF_M 4
#define WPB  4   // waves per block

__device__ __forceinline__ v8f wmma4(v2f a, v2f b, v8f c) {
  // D = A(16x4 f32) * B(4x16 f32) + C(16x16 f32)
  return __builtin_amdgcn_wmma_f32_16x16x4_f32(false, a, false, b, (short)0, c,
                                               false, false);
}

__device__ __forceinline__ float wsum(float v) {
#pragma unroll
  for (int off = 16; off >= 1; off >>= 1) v += __shfl_xor(v, off, 32);
  return v;
}

__global__ __launch_bounds__(WPB * 32) void kf_scan(
    const float* __restrict__ xg, const float* __restrict__ Fg,
    const float* __restrict__ Hg, const float* __restrict__ Qg,
    const float* __restrict__ Rg, float* __restrict__ means,
    float* __restrict__ covs) {
  const int lane = threadIdx.x & 31;
  const int wv   = threadIdx.x >> 5;
  const int b    = blockIdx.x * WPB + wv;
  const int half = lane >> 4;   // which 16-lane half
  const int l15  = lane & 15;

  __shared__ __align__(16) float ldsF[WPB][KF_S * KF_S];  // F, row-major
  __shared__ __align__(16) float ldsM[WPB][KF_S * KF_S];  // scratch, col-major
  float* Fl = ldsF[wv];
  float* Ml = ldsM[wv];

  const float* Fb = Fg + (size_t)b * KF_S * KF_S;
  const float* Hb = Hg + (size_t)b * KF_M * KF_S;
  const float* Qb = Qg + (size_t)b * KF_S * KF_S;
  const float* Rb = Rg + (size_t)b * KF_M * KF_M;
  const float* xb = xg + (size_t)b * KF_T * KF_M;

  // ---- stage F row-major into LDS (lane owns row `lane`) ----
#pragma unroll
  for (int j = 0; j < KF_S; j += 4)
    *(float4*)(Fl + lane * KF_S + j) = *(const float4*)(Fb + lane * KF_S + j);

  // ---- F fragments: A-frag(tile-row it, K-chunk kk). The f32 A(16x4) and
  //      B(4x16) lane layouts are transposes, so ff[jt*8+kk] also serves as
  //      the B-frag of F^T for X*F^T. ----
  v2f ff[16];
#pragma unroll
  for (int it = 0; it < 2; ++it)
#pragma unroll
    for (int kk = 0; kk < 8; ++kk) {
      int ad = (16 * it + l15) * KF_S + 4 * kk + 2 * half;
      v2f f; f.x = Fl[ad]; f.y = Fl[ad + 1];
      ff[it * 8 + kk] = f;
    }

  // ---- H embedded as zero-padded 16x32 A-frags (rows 4..15 = 0) ----
  v2f haf[8];
#pragma unroll
  for (int kk = 0; kk < 8; ++kk) {
    v2f h; h.x = 0.f; h.y = 0.f;
    if (l15 < 4) {
      int c = 4 * kk + 2 * half;
      h.x = Hb[l15 * KF_S + c];
      h.y = Hb[l15 * KF_S + c + 1];
    }
    haf[kk] = h;
  }
  // H columns: hcol[m] = H[m][lane]
  float hcol[4];
#pragma unroll
  for (int m = 0; m < 4; ++m) hcol[m] = Hb[m * KF_S + lane];
  // R broadcast to all lanes
  float r44[16];
#pragma unroll
  for (int i = 0; i < 16; ++i) r44[i] = Rb[i];

  // ---- Q as C/D fragments (accumulator seed for P' = FC*F^T + Q) ----
  v8f qf[2][2];
#pragma unroll
  for (int it = 0; it < 2; ++it)
#pragma unroll
    for (int jt = 0; jt < 2; ++jt)
#pragma unroll
      for (int v = 0; v < 8; ++v)
        qf[it][jt][v] = Qb[(16 * it + v + 8 * half) * KF_S + 16 * jt + l15];

  // ---- P0 = I in C/D frags and column-major in Ml ----
  v8f p[2][2];
#pragma unroll
  for (int it = 0; it < 2; ++it)
#pragma unroll
    for (int jt = 0; jt < 2; ++jt)
#pragma unroll
      for (int v = 0; v < 8; ++v)
        p[it][jt][v] = ((16 * it + v + 8 * half) == (16 * jt + l15)) ? 1.f : 0.f;
#pragma unroll
  for (int r = 0; r < KF_S; ++r) Ml[lane * KF_S + r] = (r == lane) ? 1.f : 0.f;

  float mean = 0.f;
  const v8f vzero = {0.f, 0.f, 0.f, 0.f, 0.f, 0.f, 0.f, 0.f};

#pragma unroll 1
  for (int t = 1; t < KF_T; ++t) {
    // measurement x[t-1] (uniform scalar loads)
    float xm[4];
#pragma unroll
    for (int m = 0; m < 4; ++m) xm[m] = xb[(t - 1) * KF_M + m];

    // ---- HC = Hext * P (P B-frags streamed from col-major Ml) : 16 WMMA ----
    v8f hcT[2];
#pragma unroll
    for (int jt = 0; jt < 2; ++jt) {
      v8f acc = vzero;
#pragma unroll
      for (int kk = 0; kk < 8; ++kk) {
        int ad = (16 * jt + l15) * KF_S + 4 * kk + 2 * half;
        v2f bf; bf.x = Ml[ad]; bf.y = Ml[ad + 1];
        acc = wmma4(haf[kk], bf, acc);
      }
      hcT[jt] = acc;
    }
    // extract hc[m] = HC[m][lane] from the C/D tiles
    float hc[4];
#pragma unroll
    for (int m = 0; m < 4; ++m) {
      float t0 = __shfl(hcT[0][m], l15, 32);
      float t1 = __shfl(hcT[1][m], l15, 32);
      hc[m] = half ? t1 : t0;
    }

    // ---- Smat = HC*H^T + R (wave reductions; result in every lane) ----
    float a[16];
#pragma unroll
    for (int m = 0; m < 4; ++m)
#pragma unroll
      for (int n = 0; n < 4; ++n)
        a[m * 4 + n] = wsum(hc[m] * hcol[n]) + r44[m * 4 + n];

    // ---- resid = x - H*mean ----
    float resid[4];
#pragma unroll
    for (int m = 0; m < 4; ++m) resid[m] = xm[m] - wsum(hcol[m] * mean);

    // ---- solve Smat * Kt = HC : each lane owns column `lane` of the RHS.
    //      SPD 4x4 Gaussian elimination, no pivoting. ----
    float kt0 = hc[0], kt1 = hc[1], kt2 = hc[2], kt3 = hc[3];
    {
      float inv0 = 1.f / a[0];
      float f10 = a[4] * inv0, f20 = a[8] * inv0, f30 = a[12] * inv0;
      a[5] -= f10 * a[1]; a[6] -= f10 * a[2]; a[7] -= f10 * a[3]; kt1 -= f10 * kt0;
      a[9] -= f20 * a[1]; a[10] -= f20 * a[2]; a[11] -= f20 * a[3]; kt2 -= f20 * kt0;
      a[13] -= f30 * a[1]; a[14] -= f30 * a[2]; a[15] -= f30 * a[3]; kt3 -= f30 * kt0;
      float inv1 = 1.f / a[5];
      float f21 = a[9] * inv1, f31 = a[13] * inv1;
      a[10] -= f21 * a[6]; a[11] -= f21 * a[7]; kt2 -= f21 * kt1;
      a[14] -= f31 * a[6]; a[15] -= f31 * a[7]; kt3 -= f31 * kt1;
      float inv2 = 1.f / a[10];
      float f32v = a[14] * inv2;
      a[15] -= f32v * a[11]; kt3 -= f32v * kt2;
      kt3 = kt3 / a[15];
      kt2 = (kt2 - a[11] * kt3) * inv2;
      kt1 = (kt1 - a[6] * kt2 - a[7] * kt3) * inv1;
      kt0 = (kt0 - a[1] * kt1 - a[2] * kt2 - a[3] * kt3) * inv0;
    }

    // ---- mean_u = mean + Kt^T * resid (lane owns state component `lane`) ----
    float mu = mean + kt0 * resid[0] + kt1 * resid[1] + kt2 * resid[2] +
               kt3 * resid[3];

    // ---- cov_u = P + HC^T * (-Kt): one K=4 WMMA per tile (4 WMMA) ----
    float nkt[4] = {-kt0, -kt1, -kt2, -kt3};
#pragma unroll
    for (int it = 0; it < 2; ++it)
#pragma unroll
      for (int jt = 0; jt < 2; ++jt) {
        int sA = 16 * it + l15, sB = 16 * jt + l15;
        float a0 = __shfl(hc[0], sA, 32), a1 = __shfl(hc[1], sA, 32);
        float a2 = __shfl(hc[2], sA, 32), a3 = __shfl(hc[3], sA, 32);
        float b0 = __shfl(nkt[0], sB, 32), b1 = __shfl(nkt[1], sB, 32);
        float b2 = __shfl(nkt[2], sB, 32), b3 = __shfl(nkt[3], sB, 32);
        v2f A; A.x = half ? a2 : a0; A.y = half ? a3 : a1;
        v2f Bv; Bv.x = half ? b2 : b0; Bv.y = half ? b3 : b1;
        p[it][jt] = wmma4(A, Bv, p[it][jt]);   // p now holds cov_u
      }

    // ---- cov_u -> Ml (col-major, 2x b128 per tile) ----
#pragma unroll
    for (int it = 0; it < 2; ++it)
#pragma unroll
      for (int jt = 0; jt < 2; ++jt) {
        int base = (16 * jt + l15) * KF_S + 16 * it + 8 * half;
        float4 lo = {p[it][jt][0], p[it][jt][1], p[it][jt][2], p[it][jt][3]};
        float4 hi = {p[it][jt][4], p[it][jt][5], p[it][jt][6], p[it][jt][7]};
        *(float4*)(Ml + base) = lo;
        *(float4*)(Ml + base + 4) = hi;
      }

    // ---- FC = F * cov_u : 32 WMMA (B-frags hoisted per jt) ----
    v8f fc[2][2];
#pragma unroll
    for (int jt = 0; jt < 2; ++jt) {
      v2f bf8[8];
#pragma unroll
      for (int kk = 0; kk < 8; ++kk) {
        int ad = (16 * jt + l15) * KF_S + 4 * kk + 2 * half;
        v2f bf; bf.x = Ml[ad]; bf.y = Ml[ad + 1];
        bf8[kk] = bf;
      }
#pragma unroll
      for (int it = 0; it < 2; ++it) {
        v8f acc = vzero;
#pragma unroll
        for (int kk = 0; kk < 8; ++kk) acc = wmma4(ff[it * 8 + kk], bf8[kk], acc);
        fc[it][jt] = acc;
      }
    }

    // ---- FC -> Ml (col-major) ----
#pragma unroll
    for (int it = 0; it < 2; ++it)
#pragma unroll
      for (int jt = 0; jt < 2; ++jt) {
        int base = (16 * jt + l15) * KF_S + 16 * it + 8 * half;
        float4 lo = {fc[it][jt][0], fc[it][jt][1], fc[it][jt][2], fc[it][jt][3]};
        float4 hi = {fc[it][jt][4], fc[it][jt][5], fc[it][jt][6], fc[it][jt][7]};
        *(float4*)(Ml + base) = lo;
        *(float4*)(Ml + base + 4) = hi;
      }

    // ---- P' = FC * F^T + Q : 32 WMMA (A-frags of FC hoisted per it) ----
#pragma unroll
    for (int it = 0; it < 2; ++it) {
      v2f afc[8];
#pragma unroll
      for (int kk = 0; kk < 8; ++kk) {
        v2f f;
        f.x = Ml[(4 * kk + 2 * half) * KF_S + 16 * it + l15];
        f.y = Ml[(4 * kk + 1 + 2 * half) * KF_S + 16 * it + l15];
        afc[kk] = f;
      }
#pragma unroll
      for (int jt = 0; jt < 2; ++jt) {
        v8f acc = qf[it][jt];
#pragma unroll
        for (int kk = 0; kk < 8; ++kk) acc = wmma4(afc[kk], ff[jt * 8 + kk], acc);
        p[it][jt] = acc;
      }
    }

    // ---- mean_p = F * mean_u (row of F from LDS, mean_u via shuffles) ----
    float mp = 0.f;
#pragma unroll
    for (int rq = 0; rq < 8; ++rq) {
      float4 fr = *(const float4*)(Fl + lane * KF_S + 4 * rq);
      mp += fr.x * __shfl(mu, 4 * rq + 0, 32);
      mp += fr.y * __shfl(mu, 4 * rq + 1, 32);
      mp += fr.z * __shfl(mu, 4 * rq + 2, 32);
      mp += fr.w * __shfl(mu, 4 * rq + 3, 32);
    }
    mean = mp;

    // ---- refresh Ml with P' (needed by next step's H*P) ----
#pragma unroll
    for (int it = 0; it < 2; ++it)
#pragma unroll
      for (int jt = 0; jt < 2; ++jt) {
        int base = (16 * jt + l15) * KF_S + 16 * it + 8 * half;
        float4 lo = {p[it][jt][0], p[it][jt][1], p[it][jt][2], p[it][jt][3]};
        float4 hi = {p[it][jt][4], p[it][jt][5], p[it][jt][6], p[it][jt][7]};
        *(float4*)(Ml + base) = lo;
        *(float4*)(Ml + base + 4) = hi;
      }

    // ---- emit outputs for step t ----
    size_t cb = ((size_t)b * KF_T + t) * (KF_S * KF_S);
#pragma unroll
    for (int it = 0; it < 2; ++it)
#pragma unroll
      for (int jt = 0; jt < 2; ++jt)
#pragma unroll
        for (int v = 0; v < 8; ++v)
          covs[cb + (16 * it + v + 8 * half) * KF_S + 16 * jt + l15] =
              p[it][jt][v];
    means[((size_t)b * KF_T + t) * KF_S + lane] = mean;
  }
}

// Broadcast outputs (Hs, Rs) + t=0 slices of means/covs. Pure bandwidth.
__global__ __launch_bounds__(256) void kf_bcast(
    const float* __restrict__ Hg, const float* __restrict__ Rg,
    float* __restrict__ means, float* __restrict__ covs,
    float* __restrict__ Rs, float* __restrict__ Hs) {
  const int b = blockIdx.x;
  const int tid = threadIdx.x;
  __shared__ __align__(16) float h[KF_M * KF_S];
  __shared__ __align__(16) float r[KF_M * KF_M];
  if (tid < KF_M * KF_S) h[tid] = Hg[(size_t)b * KF_M * KF_S + tid];
  if (tid < KF_M * KF_M) r[tid] = Rg[(size_t)b * KF_M * KF_M + tid];
  __syncthreads();
  const float4* h4 = (const float4*)h;
  const float4* r4 = (const float4*)r;
  float4* Hs4 = (float4*)(Hs + (size_t)b * KF_T * KF_M * KF_S);
  float4* Rs4 = (float4*)(Rs + (size_t)b * KF_T * KF_M * KF_M);
  for (int i = tid; i < KF_T * (KF_M * KF_S / 4); i += blockDim.x)
    Hs4[i] = h4[i & (KF_M * KF_S / 4 - 1)];
  for (int i = tid; i < KF_T * (KF_M * KF_M / 4); i += blockDim.x)
    Rs4[i] = r4[i & (KF_M * KF_M / 4 - 1)];
  if (tid < KF_S) means[(size_t)b * KF_T * KF_S + tid] = 0.f;
  for (int i = tid; i < KF_S * KF_S; i += blockDim.x)
    covs[(size_t)b * KF_T * KF_S * KF_S + i] =
        ((i >> 5) == (i & 31)) ? 1.f : 0.f;
}

extern "C" void kernel_launch(void* const* d_in, const int* in_sizes, int n_in,
                              void* d_out, int out_size, void* d_ws,
                              size_t ws_size, hipStream_t stream) {
  (void)in_sizes; (void)n_in; (void)out_size; (void)d_ws; (void)ws_size;
  const float* x = (const float*)d_in[0];
  const float* F = (const float*)d_in[1];
  const float* H = (const float*)d_in[2];
  const float* Q = (const float*)d_in[3];
  const float* R = (const float*)d_in[4];

  float* means = (float*)d_out;
  float* covs  = means + (size_t)KF_B * KF_T * KF_S;
  float* Rs    = covs + (size_t)KF_B * KF_T * KF_S * KF_S;
  float* Hs    = Rs + (size_t)KF_B * KF_T * KF_M * KF_M;

  kf_scan<<<KF_B / WPB, WPB * 32, 0, stream>>>(x, F, H, Q, R, means, covs);
  kf_bcast<<<KF_B, 256, 0, stream>>>(H, R, means, covs, Rs, Hs);
}